// GraphSCA3D_15599321219557
// MI455X (gfx1250) — compile-verified
//
#include <hip/hip_runtime.h>
#include <math.h>

// Problem constants (fixed by the reference)
#define NC 100000
#define EC 3200000
#define CC 256
#define RC 128
#define GC 64

typedef float v2f __attribute__((ext_vector_type(2)));
typedef float v8f __attribute__((ext_vector_type(8)));

__device__ __forceinline__ float sigmoidf_(float v) {
    return 1.0f / (1.0f + __expf(-v));
}

// ---------------------------------------------------------------------------
// 1) init: seg_sum=0, cnt=0, deg=1 (self loop)
// ---------------------------------------------------------------------------
__global__ void k_init(float* seg_sum, float* cnt, float* deg) {
    int i = blockIdx.x * blockDim.x + threadIdx.x;
    if (i < GC * CC) seg_sum[i] = 0.0f;
    if (i < GC) cnt[i] = 0.0f;
    if (i < NC) deg[i] = 1.0f;
}

// ---------------------------------------------------------------------------
// 2) per-edge degree accumulation (dst side)
// ---------------------------------------------------------------------------
__global__ void k_deg(const int* __restrict__ dst, float* deg) {
    int e = blockIdx.x * blockDim.x + threadIdx.x;
    if (e < EC) atomicAdd(&deg[dst[e]], 1.0f);
}

// ---------------------------------------------------------------------------
// 3) h[n] = dot(x[n,:], gcn_w)  — one wave32 per node, b128 loads + shfl
// ---------------------------------------------------------------------------
__global__ void k_hdot(const float* __restrict__ x, const float* __restrict__ gcn_w,
                       float* __restrict__ h) {
    int gtid = blockIdx.x * blockDim.x + threadIdx.x;
    int node = gtid >> 5;            // one wave per node (uniform per wave)
    int lane = threadIdx.x & 31;
    if (node >= NC) return;
    const float4* xr = (const float4*)(x + (size_t)node * CC);
    const float4* wr = (const float4*)gcn_w;
    float4 a0 = xr[lane * 2 + 0];
    float4 a1 = xr[lane * 2 + 1];
    float4 w0 = wr[lane * 2 + 0];
    float4 w1 = wr[lane * 2 + 1];
    float s = a0.x * w0.x + a0.y * w0.y + a0.z * w0.z + a0.w * w0.w +
              a1.x * w1.x + a1.y * w1.y + a1.z * w1.z + a1.w * w1.w;
    #pragma unroll
    for (int off = 16; off > 0; off >>= 1) s += __shfl_down(s, off, 32);
    if (lane == 0) h[node] = s;
}

// ---------------------------------------------------------------------------
// 4) segment sum over sorted batch: block = 64 threads, thread = 4 channels
//    (float4, b128 coalesced). Run-length compression: flush atomics only
//    when the (block-uniform) batch id changes.
// ---------------------------------------------------------------------------
#define CHUNK 128
__global__ void k_segsum(const float* __restrict__ x, const int* __restrict__ batch,
                         float* seg_sum, float* cnt) {
    int t = threadIdx.x;             // 0..63, owns channels 4t..4t+3
    int n0 = blockIdx.x * CHUNK;
    int n1 = n0 + CHUNK;
    if (n1 > NC) n1 = NC;
    if (n0 >= NC) return;
    const float4* x4 = (const float4*)x;
    float4 acc = {0.0f, 0.0f, 0.0f, 0.0f};
    int prev = batch[n0];
    int run = 0;
    for (int n = n0; n < n1; ++n) {
        int b = batch[n];            // uniform across block -> no divergence
        if (b != prev) {
            float* s = &seg_sum[prev * CC + 4 * t];
            atomicAdd(s + 0, acc.x);
            atomicAdd(s + 1, acc.y);
            atomicAdd(s + 2, acc.z);
            atomicAdd(s + 3, acc.w);
            if (t == 0) atomicAdd(&cnt[prev], (float)run);
            acc = {0.0f, 0.0f, 0.0f, 0.0f}; run = 0; prev = b;
        }
        float4 v = x4[(size_t)n * (CC / 4) + t];
        acc.x += v.x; acc.y += v.y; acc.z += v.z; acc.w += v.w;
        ++run;
    }
    float* s = &seg_sum[prev * CC + 4 * t];
    atomicAdd(s + 0, acc.x);
    atomicAdd(s + 1, acc.y);
    atomicAdd(s + 2, acc.z);
    atomicAdd(s + 3, acc.w);
    if (t == 0) atomicAdd(&cnt[prev], (float)run);
}

// ---------------------------------------------------------------------------
// 5) chn_mean = seg_sum / max(cnt,1)
// ---------------------------------------------------------------------------
__global__ void k_mean(const float* __restrict__ seg_sum, const float* __restrict__ cnt,
                       float* __restrict__ mean) {
    int i = blockIdx.x * blockDim.x + threadIdx.x;
    if (i < GC * CC) mean[i] = seg_sum[i] / fmaxf(cnt[i / CC], 1.0f);
}

// ---------------------------------------------------------------------------
// 6) dis = rsqrt(deg); acc = self-loop message h*dis*dis  (deg>=1 always)
// ---------------------------------------------------------------------------
__global__ void k_dis(const float* __restrict__ deg, const float* __restrict__ h,
                      float* __restrict__ dis, float* __restrict__ acc) {
    int n = blockIdx.x * blockDim.x + threadIdx.x;
    if (n < NC) {
        float r = rsqrtf(deg[n]);
        dis[n] = r;
        acc[n] = h[n] * r * r;
    }
}

// ---------------------------------------------------------------------------
// 7) per-edge messages: acc[dst] += h[src]*dis[src]*dis[dst]
// ---------------------------------------------------------------------------
__global__ void k_msg(const int* __restrict__ src, const int* __restrict__ dst,
                      const float* __restrict__ h, const float* __restrict__ dis,
                      float* acc) {
    int e = blockIdx.x * blockDim.x + threadIdx.x;
    if (e < EC) {
        int s = src[e], d = dst[e];
        atomicAdd(&acc[d], h[s] * dis[s] * dis[d]);
    }
}

// ---------------------------------------------------------------------------
// 8) MLP layer 1 with fp32 WMMA: hidden = relu(mean @ w1 + b1)
//    mean [G=64, C=256], w1 [C=256, R=128].  One wave32 per 16x16 output tile.
//    A(16x4) layout: lane = m%16 + 16*(k>=2), vgpr = k%2
//    B(4x16) layout: lane = n%16 + 16*(k>=2), vgpr = k%2
//    C(16x16): c[r] at lane l = D[r + 8*(l>=16)][l%16]
// ---------------------------------------------------------------------------
__global__ void k_mlp1(const float* __restrict__ mean, const float* __restrict__ w1,
                       const float* __restrict__ b1, float* __restrict__ hidden) {
    int tile = blockIdx.x;              // 32 tiles: (G/16)=4 x (R/16)=8
    int tm = tile >> 3, tn = tile & 7;
    int lane = threadIdx.x & 31;
    int half = lane >> 4;               // 0: k in {0,1}; 1: k in {2,3}
    int l16  = lane & 15;
    int row = tm * 16 + l16;            // A row
    int col = tn * 16 + l16;            // B col
    v8f c = {};
    #pragma unroll 4
    for (int kk = 0; kk < CC / 4; ++kk) {
        int kb = kk * 4 + half * 2;
        v2f a, b;
        a.x = mean[row * CC + kb];
        a.y = mean[row * CC + kb + 1];
        b.x = w1[(kb) * RC + col];
        b.y = w1[(kb + 1) * RC + col];
        c = __builtin_amdgcn_wmma_f32_16x16x4_f32(false, a, false, b,
                                                  (short)0, c, false, false);
    }
    #pragma unroll
    for (int r = 0; r < 8; ++r) {
        int m = tm * 16 + r + 8 * half;
        int n = tn * 16 + l16;
        hidden[m * RC + n] = fmaxf(c[r] + b1[n], 0.0f);
    }
}

// ---------------------------------------------------------------------------
// 9) MLP layer 2 with fp32 WMMA: gate = sigmoid(hidden @ w2 + b2)
//    hidden [G=64, R=128], w2 [R=128, C=256]
// ---------------------------------------------------------------------------
__global__ void k_mlp2(const float* __restrict__ hidden, const float* __restrict__ w2,
                       const float* __restrict__ b2, float* __restrict__ gate) {
    int tile = blockIdx.x;              // 64 tiles: (G/16)=4 x (C/16)=16
    int tm = tile >> 4, tn = tile & 15;
    int lane = threadIdx.x & 31;
    int half = lane >> 4;
    int l16  = lane & 15;
    int row = tm * 16 + l16;
    int col = tn * 16 + l16;
    v8f c = {};
    #pragma unroll 4
    for (int kk = 0; kk < RC / 4; ++kk) {
        int kb = kk * 4 + half * 2;
        v2f a, b;
        a.x = hidden[row * RC + kb];
        a.y = hidden[row * RC + kb + 1];
        b.x = w2[(kb) * CC + col];
        b.y = w2[(kb + 1) * CC + col];
        c = __builtin_amdgcn_wmma_f32_16x16x4_f32(false, a, false, b,
                                                  (short)0, c, false, false);
    }
    #pragma unroll
    for (int r = 0; r < 8; ++r) {
        int m = tm * 16 + r + 8 * half;
        int n = tn * 16 + l16;
        gate[m * CC + n] = sigmoidf_(c[r] + b2[n]);
    }
}

// ---------------------------------------------------------------------------
// 10) spa = sigmoid(acc + gcn_b) (in place)
// ---------------------------------------------------------------------------
__global__ void k_spa(float* acc, const float* __restrict__ gcn_b) {
    int n = blockIdx.x * blockDim.x + threadIdx.x;
    if (n < NC) acc[n] = sigmoidf_(acc[n] + gcn_b[0]);
}

// ---------------------------------------------------------------------------
// 11) fuse: out = x * (1 + spa[n] + gate[batch[n], c])  — float4 (b128) path.
//     64 float4s per node row; batch/spa broadcast amortized over 4 channels.
// ---------------------------------------------------------------------------
__global__ void k_fuse(const float* __restrict__ x, const int* __restrict__ batch,
                       const float* __restrict__ spa, const float* __restrict__ gate,
                       float* __restrict__ out) {
    long long i = (long long)blockIdx.x * blockDim.x + threadIdx.x;  // float4 index
    if (i >= (long long)NC * (CC / 4)) return;
    int n  = (int)(i >> 6);          // CC/4 = 64 float4s per node
    int c4 = (int)(i & 63);
    const float4* x4 = (const float4*)x;
    const float4* g4 = (const float4*)gate;
    float4 xv = x4[i];
    float4 gv = g4[(size_t)batch[n] * (CC / 4) + c4];
    float  s  = 1.0f + spa[n];
    float4 ov;
    ov.x = xv.x * (s + gv.x);
    ov.y = xv.y * (s + gv.y);
    ov.z = xv.z * (s + gv.z);
    ov.w = xv.w * (s + gv.w);
    ((float4*)out)[i] = ov;
}

// ---------------------------------------------------------------------------
extern "C" void kernel_launch(void* const* d_in, const int* in_sizes, int n_in,
                              void* d_out, int out_size, void* d_ws, size_t ws_size,
                              hipStream_t stream) {
    const float* x      = (const float*)d_in[0];
    const int*   batch  = (const int*)d_in[1];
    const int*   ei     = (const int*)d_in[2];   // [2, E] row-major
    const float* w1     = (const float*)d_in[4];
    const float* b1     = (const float*)d_in[5];
    const float* w2     = (const float*)d_in[6];
    const float* b2     = (const float*)d_in[7];
    const float* gcn_w  = (const float*)d_in[8];
    const float* gcn_b  = (const float*)d_in[9];
    float* out = (float*)d_out;

    const int* src = ei;
    const int* dst = ei + EC;

    // workspace layout (floats); float4-aligned chunks
    float* ws      = (float*)d_ws;
    float* seg_sum = ws;                       // G*C = 16384
    float* cnt     = seg_sum + GC * CC;        // G   = 64
    float* mean    = cnt + GC;                 // G*C = 16384
    float* hidden  = mean + GC * CC;           // G*R = 8192
    float* gate    = hidden + GC * RC;         // G*C = 16384
    float* h       = gate + GC * CC;           // N
    float* deg     = h + NC;                   // N
    float* dis     = deg + NC;                 // N
    float* acc     = dis + NC;                 // N  (becomes spa)

    const int T = 256;
    k_init  <<<(NC + T - 1) / T, T, 0, stream>>>(seg_sum, cnt, deg);
    k_deg   <<<(EC + T - 1) / T, T, 0, stream>>>(dst, deg);
    k_hdot  <<<(NC * 32 + T - 1) / T, T, 0, stream>>>(x, gcn_w, h);
    k_segsum<<<(NC + CHUNK - 1) / CHUNK, 64, 0, stream>>>(x, batch, seg_sum, cnt);
    k_mean  <<<(GC * CC + T - 1) / T, T, 0, stream>>>(seg_sum, cnt, mean);
    k_dis   <<<(NC + T - 1) / T, T, 0, stream>>>(deg, h, dis, acc);
    k_msg   <<<(EC + T - 1) / T, T, 0, stream>>>(src, dst, h, dis, acc);
    k_mlp1  <<<(GC / 16) * (RC / 16), 32, 0, stream>>>(mean, w1, b1, hidden);
    k_mlp2  <<<(GC / 16) * (CC / 16), 32, 0, stream>>>(hidden, w2, b2, gate);
    k_spa   <<<(NC + T - 1) / T, T, 0, stream>>>(acc, gcn_b);
    {
        long long total = (long long)NC * (CC / 4);
        int blocks = (int)((total + T - 1) / T);
        k_fuse<<<blocks, T, 0, stream>>>(x, batch, acc, gate, out);
    }
}